// FPS_kNN_6012954214669
// MI455X (gfx1250) — compile-verified
//
#include <hip/hip_runtime.h>
#include <hip/hip_bf16.h>
#include <float.h>

#define B_ 8
#define N_ 8192
#define C_ 128
#define G_ 1024
#define K_ 32

typedef __attribute__((ext_vector_type(2))) float v2f;
typedef __attribute__((ext_vector_type(8))) float v8f;

// ---------------------------------------------------------------------------
// Kernel 1: farthest point sampling. One 1024-thread block per batch; each
// thread owns 8 points in registers. Serial over G steps with wave32 shuffle
// argmax + cross-wave LDS reduction. Tie-break = lowest index (jnp.argmax).
// ---------------------------------------------------------------------------
__global__ __launch_bounds__(1024) void fps_kernel(const float* __restrict__ xyz,
                                                   int* __restrict__ fps_idx)
{
    const int b   = blockIdx.x;
    const int tid = threadIdx.x;
    const float* base = xyz + (size_t)b * N_ * 3;

    float px[8], py[8], pz[8], dmin[8];
#pragma unroll
    for (int j = 0; j < 8; ++j) {
        int p = tid + j * 1024;
        px[j] = base[p * 3 + 0];
        py[j] = base[p * 3 + 1];
        pz[j] = base[p * 3 + 2];
        dmin[j] = FLT_MAX;
    }

    __shared__ float s_best[32];
    __shared__ int   s_bidx[32];
    __shared__ float s_cx, s_cy, s_cz;
    __shared__ int   s_sel;

    if (tid == 0) { s_sel = 0; s_cx = base[0]; s_cy = base[1]; s_cz = base[2]; }
    __syncthreads();

    int* out = fps_idx + (size_t)b * G_;
    for (int s = 0; s < G_; ++s) {
        int   last = s_sel;
        float cx = s_cx, cy = s_cy, cz = s_cz;
        if (tid == 0) out[s] = last;   // reference emits 'last' before update

        float bd = -1.0f; int bi = 0x7fffffff;
#pragma unroll
        for (int j = 0; j < 8; ++j) {
            float dx = px[j] - cx, dy = py[j] - cy, dz = pz[j] - cz;
            float d = dx * dx + dy * dy + dz * dz;
            dmin[j] = fminf(dmin[j], d);
            int p = tid + j * 1024;
            if (dmin[j] > bd || (dmin[j] == bd && p < bi)) { bd = dmin[j]; bi = p; }
        }
        // wave32 argmax reduction
#pragma unroll
        for (int off = 16; off > 0; off >>= 1) {
            float od = __shfl_xor(bd, off, 32);
            int   oi = __shfl_xor(bi, off, 32);
            if (od > bd || (od == bd && oi < bi)) { bd = od; bi = oi; }
        }
        if ((tid & 31) == 0) { s_best[tid >> 5] = bd; s_bidx[tid >> 5] = bi; }
        __syncthreads();
        if (tid < 32) {
            float bd2 = s_best[tid]; int bi2 = s_bidx[tid];
#pragma unroll
            for (int off = 16; off > 0; off >>= 1) {
                float od = __shfl_xor(bd2, off, 32);
                int   oi = __shfl_xor(bi2, off, 32);
                if (od > bd2 || (od == bd2 && oi < bi2)) { bd2 = od; bi2 = oi; }
            }
            if (tid == 0) s_sel = bi2;
        }
        __syncthreads();
        int sel = s_sel;
        if (tid == (sel & 1023)) {   // owning thread broadcasts coords from registers
            int j = sel >> 10;
            float cx2 = px[0], cy2 = py[0], cz2 = pz[0];
#pragma unroll
            for (int jj = 1; jj < 8; ++jj)
                if (jj == j) { cx2 = px[jj]; cy2 = py[jj]; cz2 = pz[jj]; }
            s_cx = cx2; s_cy = cy2; s_cz = cz2;
        }
        __syncthreads();
    }
}

// ---------------------------------------------------------------------------
// Kernel 2: per-point squared norms (B-fragment's 4th row for the WMMA trick).
// ---------------------------------------------------------------------------
__global__ void sqnorm_kernel(const float* __restrict__ xyz, float* __restrict__ sq)
{
    int i = blockIdx.x * blockDim.x + threadIdx.x;   // flat b*N + n
    if (i < B_ * N_) {
        float x = xyz[i * 3 + 0], y = xyz[i * 3 + 1], z = xyz[i * 3 + 2];
        sq[i] = x * x + y * y + z * z;
    }
}

// ---------------------------------------------------------------------------
// Kernel 3: gather lc_* (one wave per sampled center; 128-ch rows as float4).
// ---------------------------------------------------------------------------
__global__ void lc_gather_kernel(const float* __restrict__ xyz, const float* __restrict__ x,
                                 const float* __restrict__ rgb, const float* __restrict__ rgbx,
                                 const int* __restrict__ fps_idx,
                                 float* __restrict__ lc_xyz, float* __restrict__ lc_x,
                                 float* __restrict__ lc_rgb, float* __restrict__ lc_rgbx)
{
    int gid = blockIdx.x * blockDim.x + threadIdx.x;
    int row = gid >> 5;
    int lane = gid & 31;
    if (row >= B_ * G_) return;
    int b = row >> 10;                    // G_ == 1024
    int idx = fps_idx[row];
    size_t src = (size_t)b * N_ + idx;
    if (lane < 3)       lc_xyz[(size_t)row * 3 + lane] = xyz[src * 3 + lane];
    else if (lane < 6)  lc_rgb[(size_t)row * 3 + lane - 3] = rgb[src * 3 + lane - 3];
    const float4* xs = (const float4*)(x + src * C_);
    float4*       xd = (float4*)(lc_x + (size_t)row * C_);
    xd[lane] = xs[lane];                  // 32 lanes * 4 = 128 channels
    const float4* rs = (const float4*)(rgbx + src * C_);
    float4*       rd = (float4*)(lc_rgbx + (size_t)row * C_);
    rd[lane] = rs[lane];
}

// ---------------------------------------------------------------------------
// Kernel 4: kNN via V_WMMA_F32_16X16X4_F32.
// One wave per 16-query tile. A row m = [-2qx,-2qy,-2qz, 1]; B col n =
// [px,py,pz,||p||^2] => D[m][n] = ||p||^2 - 2 q.p (reference's rank metric).
// Distances staged in padded LDS; lanes 0-15 each maintain a top-32 list.
// ---------------------------------------------------------------------------
__global__ __launch_bounds__(32) void knn_select_kernel(const float* __restrict__ xyz,
                                                        const float* __restrict__ sq,
                                                        const float* __restrict__ lc_xyz,
                                                        int* __restrict__ knn_idx)
{
    const int b    = blockIdx.x >> 6;   // 64 query tiles per batch (G/16)
    const int qt   = blockIdx.x & 63;
    const int g0   = qt * 16;
    const int lane = threadIdx.x;
    const int col  = lane & 15;
    const bool hi  = lane >= 16;

    __shared__ float sdist[16][132];    // pad 128->132 dwords: no bank conflicts

    // A fragment (16x4 f32): lanes 0-15 hold K=0/1, lanes 16-31 hold K=2/3.
    const float* q = lc_xyz + ((size_t)b * G_ + g0 + col) * 3;
    float qx = q[0], qy = q[1], qz = q[2];
    v2f a;
    a.x = hi ? (-2.0f * qz) : (-2.0f * qx);
    a.y = hi ? 1.0f         : (-2.0f * qy);

    float bestd[K_]; int besti[K_];
#pragma unroll
    for (int j = 0; j < K_; ++j) { bestd[j] = FLT_MAX; besti[j] = 0; }
    float worst = FLT_MAX; int wpos = 0;

    const float* pbase = xyz + (size_t)b * N_ * 3;
    const float* sbase = sq + (size_t)b * N_;

    for (int n0 = 0; n0 < N_; n0 += 128) {
#pragma unroll
        for (int t = 0; t < 8; ++t) {
            int n = n0 + t * 16 + col;
            float p0 = pbase[n * 3 + 0];
            float p1 = pbase[n * 3 + 1];
            float p2 = pbase[n * 3 + 2];
            float sn = sbase[n];
            if (n0 + 128 < N_) __builtin_prefetch(pbase + (size_t)(n + 128) * 3, 0, 1);
            // B fragment (4x16 f32): lanes 0-15 hold K=0/1 rows, 16-31 hold K=2/3.
            v2f bf;
            bf.x = hi ? p2 : p0;
            bf.y = hi ? sn : p1;
            v8f c = {};
            v8f d = __builtin_amdgcn_wmma_f32_16x16x4_f32(
                false, a, false, bf, (short)0, c, false, false);
            // D layout: VGPR r -> M = r (lanes 0-15) or r+8 (lanes 16-31), N = col
#pragma unroll
            for (int r = 0; r < 8; ++r) {
                int m = hi ? (r + 8) : r;
                sdist[m][t * 16 + col] = d[r];
            }
        }
        __syncthreads();
        if (lane < 16) {
#pragma unroll 1
            for (int j = 0; j < 128; ++j) {
                float dd = sdist[lane][j];
                if (dd < worst) {                 // strict < keeps earliest index on ties
                    bestd[wpos] = dd; besti[wpos] = n0 + j;
                    float w = -FLT_MAX; int wp = 0;
#pragma unroll
                    for (int jj = 0; jj < K_; ++jj)
                        if (bestd[jj] > w) { w = bestd[jj]; wp = jj; }
                    worst = w; wpos = wp;
                }
            }
        }
        __syncthreads();
    }

    // Sort ascending by distance (top_k returns nearest-first), then emit.
    if (lane < 16) {
        int* out = knn_idx + ((size_t)b * G_ + g0 + lane) * K_;
#pragma unroll 1
        for (int i = 0; i < K_; ++i) {
            float mv = bestd[i]; int mp = i;
#pragma unroll 1
            for (int j = i + 1; j < K_; ++j)
                if (bestd[j] < mv || (bestd[j] == mv && besti[j] < besti[mp])) {
                    mv = bestd[j]; mp = j;
                }
            float td = bestd[i]; bestd[i] = bestd[mp]; bestd[mp] = td;
            int   ti = besti[i]; besti[i] = besti[mp]; besti[mp] = ti;
            out[i] = besti[i];
        }
    }
}

// ---------------------------------------------------------------------------
// Kernel 5: gather knn_* (one wave per (b,g,k) neighbor row).
// ---------------------------------------------------------------------------
__global__ void knn_gather_kernel(const float* __restrict__ xyz, const float* __restrict__ x,
                                  const float* __restrict__ rgb, const float* __restrict__ rgbx,
                                  const int* __restrict__ knn_idx,
                                  float* __restrict__ knn_xyz, float* __restrict__ knn_x,
                                  float* __restrict__ knn_rgb, float* __restrict__ knn_rgbx)
{
    int gid = blockIdx.x * blockDim.x + threadIdx.x;
    int row = gid >> 5;                  // B*G*K rows
    int lane = gid & 31;
    if (row >= B_ * G_ * K_) return;
    int b = row / (G_ * K_);
    int idx = knn_idx[row];
    size_t src = (size_t)b * N_ + idx;
    if (lane < 3)       knn_xyz[(size_t)row * 3 + lane] = xyz[src * 3 + lane];
    else if (lane < 6)  knn_rgb[(size_t)row * 3 + lane - 3] = rgb[src * 3 + lane - 3];
    const float4* xs = (const float4*)(x + src * C_);
    float4*       xd = (float4*)(knn_x + (size_t)row * C_);
    xd[lane] = xs[lane];
    const float4* rs = (const float4*)(rgbx + src * C_);
    float4*       rd = (float4*)(knn_rgbx + (size_t)row * C_);
    rd[lane] = rs[lane];
}

extern "C" void kernel_launch(void* const* d_in, const int* in_sizes, int n_in,
                              void* d_out, int out_size, void* d_ws, size_t ws_size,
                              hipStream_t stream)
{
    (void)in_sizes; (void)n_in; (void)out_size; (void)ws_size;
    const float* xyz  = (const float*)d_in[0];
    const float* x    = (const float*)d_in[1];
    const float* rgb  = (const float*)d_in[2];
    const float* rgbx = (const float*)d_in[3];
    // d_in[4]=group_num, d_in[5]=k_neighbors are device scalars; shapes are
    // fixed by setup_inputs (G=1024, K=32) and cannot be read during capture.

    float* out      = (float*)d_out;
    float* lc_xyz   = out;
    float* lc_x     = lc_xyz   + (size_t)B_ * G_ * 3;
    float* lc_rgb   = lc_x     + (size_t)B_ * G_ * C_;
    float* lc_rgbx  = lc_rgb   + (size_t)B_ * G_ * 3;
    float* knn_xyz  = lc_rgbx  + (size_t)B_ * G_ * C_;
    float* knn_x    = knn_xyz  + (size_t)B_ * G_ * K_ * 3;
    float* knn_rgb  = knn_x    + (size_t)B_ * G_ * K_ * C_;
    float* knn_rgbx = knn_rgb  + (size_t)B_ * G_ * K_ * 3;

    int*   ws_fps = (int*)d_ws;                       // B*G
    int*   ws_knn = ws_fps + B_ * G_;                 // B*G*K
    float* ws_sq  = (float*)(ws_knn + B_ * G_ * K_);  // B*N

    fps_kernel<<<B_, 1024, 0, stream>>>(xyz, ws_fps);
    sqnorm_kernel<<<(B_ * N_ + 255) / 256, 256, 0, stream>>>(xyz, ws_sq);
    lc_gather_kernel<<<(B_ * G_ * 32 + 255) / 256, 256, 0, stream>>>(
        xyz, x, rgb, rgbx, ws_fps, lc_xyz, lc_x, lc_rgb, lc_rgbx);
    knn_select_kernel<<<B_ * (G_ / 16), 32, 0, stream>>>(xyz, ws_sq, lc_xyz, ws_knn);
    knn_gather_kernel<<<(B_ * G_ * K_ * 32 + 255) / 256, 256, 0, stream>>>(
        xyz, x, rgb, rgbx, ws_knn, knn_xyz, knn_x, knn_rgb, knn_rgbx);
}